// Speller_72748156060303
// MI455X (gfx1250) — compile-verified
//
#include <hip/hip_runtime.h>
#include <math.h>

#define B_    64
#define T_    1024
#define HC_   512
#define HS_   512
#define V_    256
#define DIN_  768     // V + HC
#define NG_   2048    // 4*HS
#define SMAX_ 100

typedef __attribute__((ext_vector_type(2))) float v2f;
typedef __attribute__((ext_vector_type(8))) float v8f;

__device__ __forceinline__ float sigm_(float x) { return 1.0f / (1.0f + __expf(-x)); }

// ---------------------------------------------------------------- init state
__global__ void k_init(const float* __restrict__ h, float* __restrict__ x,
                       float* __restrict__ cp, float* __restrict__ sc,
                       int* __restrict__ eos) {
  int idx = blockIdx.x * 256 + threadIdx.x;
  if (idx < B_ * DIN_) {
    int b = idx / DIN_, j = idx % DIN_;
    float v;
    if (j < V_) v = (j == V_ - 2) ? 1.0f : 0.0f;        // one-hot SOS
    else        v = h[(size_t)b * T_ * HC_ + (j - V_)]; // h[:,0,:]
    x[idx] = v;
  }
  if (idx < B_ * HS_) {
    cp[idx] = 0.0f;
    int b = idx >> 9, d = idx & 511;
    sc[b * 1024 + d] = 0.0f;                            // hp0 = 0 (s slot)
  }
  if (idx < B_) eos[idx] = SMAX_;
}

// ---------------------------------------------------------------- layernorm
__global__ void k_layernorm(const float* __restrict__ x, float* __restrict__ xn) {
  __shared__ float red[256];
  int b = blockIdx.x, tid = threadIdx.x;
  const float* xr = x + b * DIN_;
  float v0 = xr[tid], v1 = xr[tid + 256], v2 = xr[tid + 512];
  red[tid] = v0 + v1 + v2;
  __syncthreads();
  for (int o = 128; o > 0; o >>= 1) { if (tid < o) red[tid] += red[tid + o]; __syncthreads(); }
  float mu = red[0] * (1.0f / DIN_);
  __syncthreads();
  float d0 = v0 - mu, d1 = v1 - mu, d2 = v2 - mu;
  red[tid] = d0 * d0 + d1 * d1 + d2 * d2;
  __syncthreads();
  for (int o = 128; o > 0; o >>= 1) { if (tid < o) red[tid] += red[tid + o]; __syncthreads(); }
  float r = rsqrtf(red[0] * (1.0f / DIN_) + 1e-5f);
  float* o = xn + b * DIN_;
  o[tid] = d0 * r; o[tid + 256] = d1 * r; o[tid + 512] = d2 * r;
}

// ------------------------------------------------- WMMA f32 16x16x4 GEMM
// One wave computes NBLK adjacent 16x16 tiles, reusing the A fragment across
// NBLK B fragments. K-loop is software-pipelined: loads for step k+4 are
// issued (into distinct registers) before the WMMAs of step k, so NBLK+1
// loads stay in flight behind the matrix ops.
// A-frag: lane holds A[lane&15][k + 2*(lane>>4) + {0,1}]            (float2)
// B-frag (4x16 W^T tile): lane holds W[n+(lane&15)][k + 2*(lane>>4) + {0,1}]
// C/D: VGPR r -> row m0 + r + 8*(lane>>4), col n + (lane&15)
template <int NBLK>
__device__ __forceinline__ void wmma_seg(const float* __restrict__ arow,
                                         const float* __restrict__ wrow,
                                         size_t wjstride, int K, v8f* c) {
  v2f a_cur = *(const v2f*)(arow);
  v2f w_cur[NBLK];
  #pragma unroll
  for (int j = 0; j < NBLK; ++j) w_cur[j] = *(const v2f*)(wrow + j * wjstride);

  #pragma unroll 2
  for (int k = 0; k < K - 4; k += 4) {
    v2f a_nxt = *(const v2f*)(arow + k + 4);
    v2f w_nxt[NBLK];
    #pragma unroll
    for (int j = 0; j < NBLK; ++j)
      w_nxt[j] = *(const v2f*)(wrow + j * wjstride + k + 4);
    #pragma unroll
    for (int j = 0; j < NBLK; ++j)
      c[j] = __builtin_amdgcn_wmma_f32_16x16x4_f32(false, a_cur, false, w_cur[j],
                                                   (short)0, c[j], false, false);
    a_cur = a_nxt;
    #pragma unroll
    for (int j = 0; j < NBLK; ++j) w_cur[j] = w_nxt[j];
  }
  #pragma unroll
  for (int j = 0; j < NBLK; ++j)
    c[j] = __builtin_amdgcn_wmma_f32_16x16x4_f32(false, a_cur, false, w_cur[j],
                                                 (short)0, c[j], false, false);
}

template <int NBLK>
__global__ void k_wmma_gemm(const float* __restrict__ A1, int lda1, int K1,
                            const float* __restrict__ W1,
                            const float* __restrict__ A2, int lda2, int K2,
                            const float* __restrict__ W2,
                            const float* __restrict__ bias1,
                            const float* __restrict__ bias2,
                            float* __restrict__ out, int ldo, int N) {
  int wave = threadIdx.x >> 5;
  int lane = threadIdx.x & 31;
  int task = blockIdx.x * (blockDim.x >> 5) + wave;
  int n_groups = N / (16 * NBLK);
  int m0 = (task / n_groups) << 4;
  int n0 = (task % n_groups) * (16 * NBLK);
  int half = lane >> 4;
  int ml = lane & 15;

  v8f c[NBLK];
  #pragma unroll
  for (int j = 0; j < NBLK; ++j) c[j] = (v8f){};

  wmma_seg<NBLK>(A1 + (size_t)(m0 + ml) * lda1 + 2 * half,
                 W1 + (size_t)(n0 + ml) * K1 + 2 * half,
                 (size_t)16 * K1, K1, c);
  if (A2)
    wmma_seg<NBLK>(A2 + (size_t)(m0 + ml) * lda2 + 2 * half,
                   W2 + (size_t)(n0 + ml) * K2 + 2 * half,
                   (size_t)16 * K2, K2, c);

  #pragma unroll
  for (int j = 0; j < NBLK; ++j) {
    int n = n0 + 16 * j + ml;
    float bias = 0.0f;
    if (bias1) bias += bias1[n];
    if (bias2) bias += bias2[n];
    #pragma unroll
    for (int r = 0; r < 8; ++r) {
      int m = m0 + r + 8 * half;
      out[(size_t)m * ldo + n] = c[j][r] + bias;
    }
  }
}

// ---------------------------------------------------------------- LSTM cell
__global__ void k_lstm(const float* __restrict__ gates, float* __restrict__ cp,
                       float* __restrict__ sc) {
  int idx = blockIdx.x * 256 + threadIdx.x;   // B*HS threads
  int b = idx >> 9, d = idx & 511;
  const float* g = gates + (size_t)b * NG_;
  float ig = sigm_(g[d]);
  float fg = sigm_(g[HS_ + d]);
  float gg = tanhf(g[2 * HS_ + d]);
  float og = sigm_(g[3 * HS_ + d]);
  float cc = fg * cp[idx] + ig * gg;
  cp[idx] = cc;
  sc[b * 1024 + d] = og * tanhf(cc);          // s -> first half of concat buf
}

// ---------------------------------------------------- attention scores (GEMV)
__global__ void k_attn_scores(const float* __restrict__ sc,
                              const float* __restrict__ h,
                              float* __restrict__ e) {
  __shared__ float s_lds[HS_];
  int wave = threadIdx.x >> 5, lane = threadIdx.x & 31;
  int row = blockIdx.x * 8 + wave;            // row in [0, B*T)
  int b = row >> 10, t = row & 1023;
  for (int i = threadIdx.x; i < HS_; i += 256) s_lds[i] = sc[b * 1024 + i];
  __syncthreads();
  const float* hr = h + (size_t)(b * T_ + t) * HC_;
  float acc = 0.0f;
  #pragma unroll
  for (int k = lane; k < HC_; k += 32) acc += s_lds[k] * hr[k];
  for (int o = 16; o > 0; o >>= 1) acc += __shfl_xor(acc, o, 32);
  if (lane == 0) e[row] = acc;
}

// ---------------------------------------------------------------- softmax(att)
__global__ void k_attn_softmax(const float* __restrict__ e, float* __restrict__ a,
                               float* __restrict__ att_out) {
  __shared__ float red[256];
  int b = blockIdx.x, tid = threadIdx.x;
  const float* er = e + b * T_;
  float v[4];
  float mx = -3.0e38f;
  #pragma unroll
  for (int i = 0; i < 4; ++i) { v[i] = er[tid + 256 * i]; mx = fmaxf(mx, v[i]); }
  red[tid] = mx; __syncthreads();
  for (int o = 128; o > 0; o >>= 1) { if (tid < o) red[tid] = fmaxf(red[tid], red[tid + o]); __syncthreads(); }
  mx = red[0]; __syncthreads();
  float sum = 0.0f;
  #pragma unroll
  for (int i = 0; i < 4; ++i) { v[i] = __expf(v[i] - mx); sum += v[i]; }
  red[tid] = sum; __syncthreads();
  for (int o = 128; o > 0; o >>= 1) { if (tid < o) red[tid] += red[tid + o]; __syncthreads(); }
  float inv = 1.0f / red[0];
  #pragma unroll
  for (int i = 0; i < 4; ++i) {
    float av = v[i] * inv;
    a[b * T_ + tid + 256 * i] = av;
    att_out[b * T_ + tid + 256 * i] = av;
  }
}

// ---------------------------------------------------------------- context GEMV
__global__ void k_ctx(const float* __restrict__ a, const float* __restrict__ h,
                      float* __restrict__ sc, float* __restrict__ x) {
  __shared__ float a_lds[T_];
  int b = blockIdx.y;
  int d = blockIdx.x * 256 + threadIdx.x;     // 0..511 over 2 blocks
  for (int i = threadIdx.x; i < T_; i += 256) a_lds[i] = a[b * T_ + i];
  __syncthreads();
  const float* hb = h + (size_t)b * T_ * HC_ + d;
  float acc = 0.0f;
  #pragma unroll 8
  for (int t = 0; t < T_; ++t) acc += a_lds[t] * hb[(size_t)t * HC_];
  sc[b * 1024 + 512 + d] = acc;               // ctx -> second half of concat buf
  x[b * DIN_ + V_ + d] = acc;                 // x_next[:, 256:768] = ctx
}

// --------------------------------------------- output softmax + argmax/eos
__global__ void k_out_softmax(const float* __restrict__ logits,
                              float* __restrict__ yhat, float* __restrict__ x,
                              int* __restrict__ eos, int st) {
  __shared__ float red[256];
  __shared__ int  redi[256];
  int b = blockIdx.x, v = threadIdx.x;
  float l = logits[b * V_ + v];
  red[v] = l; redi[v] = v; __syncthreads();
  for (int o = 128; o > 0; o >>= 1) {
    if (v < o && red[v + o] > red[v]) { red[v] = red[v + o]; redi[v] = redi[v + o]; }
    __syncthreads();
  }
  float mx = red[0];
  int am = redi[0];
  __syncthreads();
  float ex = __expf(l - mx);
  red[v] = ex; __syncthreads();
  for (int o = 128; o > 0; o >>= 1) { if (v < o) red[v] += red[v + o]; __syncthreads(); }
  float y = ex / red[0];
  yhat[((size_t)b * SMAX_ + st) * V_ + v] = y;   // y_hats[b][st][v]
  x[b * DIN_ + v] = y;                           // x_next[:, 0:256] = y
  if (v == 0 && am == V_ - 1) atomicMin(&eos[b], st);
}

__global__ void k_seqlen(const int* __restrict__ eos, int* __restrict__ out) {
  int b = threadIdx.x;
  if (b < B_) out[b] = eos[b];
}

// ================================================================== launcher
extern "C" void kernel_launch(void* const* d_in, const int* in_sizes, int n_in,
                              void* d_out, int out_size, void* d_ws, size_t ws_size,
                              hipStream_t stream) {
  const float* h     = (const float*)d_in[0];
  const float* W_ih  = (const float*)d_in[1];
  const float* W_hh  = (const float*)d_in[2];
  const float* b_ih  = (const float*)d_in[3];
  const float* b_hh  = (const float*)d_in[4];
  const float* W_out = (const float*)d_in[5];
  const float* b_out = (const float*)d_in[6];

  float* outf = (float*)d_out;
  float* yhat = outf;                                   // 64*100*256
  int*   seq  = (int*)(outf + (size_t)B_ * SMAX_ * V_); // 64 (int bits)
  float* att  = outf + (size_t)B_ * SMAX_ * V_ + B_;    // 100*64*1024

  float* ws = (float*)d_ws;
  float* x      = ws;                 // 64*768
  float* xn     = x + B_ * DIN_;      // 64*768
  float* gates  = xn + B_ * DIN_;     // 64*2048
  float* sc     = gates + B_ * NG_;   // 64*1024  [s | ctx]
  float* cp     = sc + B_ * 1024;     // 64*512
  float* e      = cp + B_ * HS_;      // 64*1024
  float* a      = e + B_ * T_;        // 64*1024
  float* logits = a + B_ * T_;        // 64*256
  int*   eos    = (int*)(logits + B_ * V_);

  k_init<<<(B_ * DIN_ + 255) / 256, 256, 0, stream>>>(h, x, cp, sc, eos);

  for (int st = 0; st < SMAX_; ++st) {
    k_layernorm<<<B_, 256, 0, stream>>>(x, xn);
    // gates = xn @ W_ih^T + hp @ W_hh^T + b_ih + b_hh   (hp = sc[:, 0:512])
    // NBLK=4: tasks = 4 * (2048/64) = 128 -> 16 blocks of 8 waves
    k_wmma_gemm<4><<<16, 256, 0, stream>>>(xn, DIN_, DIN_, W_ih,
                                           sc, 1024, HS_, W_hh,
                                           b_ih, b_hh, gates, NG_, NG_);
    k_lstm<<<(B_ * HS_) / 256, 256, 0, stream>>>(gates, cp, sc);
    k_attn_scores<<<(B_ * T_) / 8, 256, 0, stream>>>(sc, h, e);
    k_attn_softmax<<<B_, 256, 0, stream>>>(e, a, att + (size_t)st * B_ * T_);
    k_ctx<<<dim3(2, B_), 256, 0, stream>>>(a, h, sc, x);
    // logits = [s|ctx] @ W_out^T + b_out
    // NBLK=2: tasks = 4 * (256/32) = 32 -> 4 blocks of 8 waves
    k_wmma_gemm<2><<<4, 256, 0, stream>>>(sc, 1024, 1024, W_out,
                                          nullptr, 0, 0, nullptr,
                                          b_out, nullptr, logits, V_, V_);
    k_out_softmax<<<B_, 256, 0, stream>>>(logits, yhat, x, eos, st);
  }
  k_seqlen<<<1, 64, 0, stream>>>(eos, seq);
}